// EllipsoidRenderer_14070312862345
// MI455X (gfx1250) — compile-verified
//
#include <hip/hip_runtime.h>

// ============================================================================
// Ellipsoid volume renderer for MI455X (gfx1250, wave32).
//  - Intersection quadratics (a,b,c) for all 128 ellipsoids x 16 rays per block
//    computed as coefficient x feature GEMMs with V_WMMA_F32_16X16X4_F32.
//  - Per-ray 256-event branchless bitonic sort of packed 64-bit
//    (orderedFloatKey<<32 | payload) records in LDS (16 rays in parallel).
//  - Volumetric integration via 3 LDS prefix/suffix scans (color-cumsum folded
//    into a suffix scan by swapping the double sum).
//  - Fast-path math: v_rcp_f32 / v_sqrt_f32 (denominators positive, clamped).
// ============================================================================

typedef float v2f __attribute__((ext_vector_type(2)));
typedef float v8f __attribute__((ext_vector_type(8)));
typedef unsigned long long u64;

#define N_ELL 128
#define NEV   256          // 2 * N_ELL events per ray
#define RPB   16           // rays per block
#define NTHR  256

// Order-preserving float -> u32 map (total order matches float <).
__device__ __forceinline__ unsigned okey(float f) {
  const unsigned b = __float_as_uint(f);
  return (b & 0x80000000u) ? ~b : (b | 0x80000000u);
}
__device__ __forceinline__ float unokey(unsigned k) {
  const unsigned b = (k & 0x80000000u) ? (k ^ 0x80000000u) : ~k;
  return __uint_as_float(b);
}

// One K=4 slice of D = A(16xK) x B(Kx16) + C.
// f32 WMMA operand layout (ISA 7.12.2):
//   A: M = lane%16, K = vgpr + 2*(lane/16)
//   B: N = lane%16, K = vgpr + 2*(lane/16)
//   C: M = vgpr + 8*(lane/16), N = lane%16
__device__ __forceinline__ v8f wmma_chunk(const float* __restrict__ coef, int KP,
                                          const float* __restrict__ feat,
                                          int rowBase, int row, int kh,
                                          int q, v8f acc) {
  const int kbase = 4 * q + 2 * kh;
  v2f a, b;
  a.x = coef[(rowBase + row) * KP + kbase + 0];
  a.y = coef[(rowBase + row) * KP + kbase + 1];
  b.x = feat[(kbase + 0) * RPB + row];
  b.y = feat[(kbase + 1) * RPB + row];
  return __builtin_amdgcn_wmma_f32_16x16x4_f32(false, a, false, b,
                                               (short)0, acc, false, false);
}

__global__ __launch_bounds__(NTHR)
void ellipsoid_render_kernel(const float* __restrict__ means3D,
                             const float* __restrict__ rays_o,
                             const float* __restrict__ rays_d,
                             const float* __restrict__ rotations,
                             const float* __restrict__ scales,
                             const float* __restrict__ colors,
                             const float* __restrict__ opacities,
                             float* __restrict__ out)
{
  // ---- LDS layout (15360 floats = 60 KB) ----
  __shared__ __align__(16) float smem[15360];
  float* cA    = smem;            // [128][ 8]  a-coefficients      (phase 1)
  float* cB    = smem + 1024;     // [128][12]  b-coefficients      (phase 1)
  float* cC    = smem + 2560;     // [128][12]  c-coefficients      (phase 1)
  float* dens  = smem;            // [16][256]  density (ALIAS of coef region)
  float* fA    = smem + 4096;     // [ 8][16]   ray features for a
  float* fB    = smem + 4224;     // [12][16]   ray features for b
  float* fC    = smem + 4416;     // [12][16]   ray features for c
  float* colOp = smem + 4608;     // [128][4]   color.xyz, opacity
  u64*   evK   = (u64*)(smem + 5120);   // [16][256] packed events (phases 1-2)
  float* evT   = smem + 5120;     // [16][256]  times, ALIAS low half of evK
  float* G     = smem + 9216;     // [16][256]  scans, ALIAS high half of evK
  unsigned short* evP = (unsigned short*)(smem + 13312); // [16][256] (id<<2)|(io+1)

  const int tid = threadIdx.x;

  // ======================= Phase 0: per-block setup =========================
  // Per-ellipsoid quadratic-form coefficients: Q = T^T T, T = diag(1/s) R^T.
  if (tid < N_ELL) {
    const int e = tid;
    float R_[3][3], T_[3][3];
    #pragma unroll
    for (int i = 0; i < 3; ++i)
      #pragma unroll
      for (int j = 0; j < 3; ++j)
        R_[i][j] = rotations[e*9 + i*3 + j];
    const float inv_s[3] = {__builtin_amdgcn_rcpf(scales[e*3+0]),
                            __builtin_amdgcn_rcpf(scales[e*3+1]),
                            __builtin_amdgcn_rcpf(scales[e*3+2])};
    #pragma unroll
    for (int i = 0; i < 3; ++i)
      #pragma unroll
      for (int j = 0; j < 3; ++j)
        T_[i][j] = R_[j][i] * inv_s[i];
    float Q[3][3];
    #pragma unroll
    for (int i = 0; i < 3; ++i)
      #pragma unroll
      for (int j = 0; j < 3; ++j)
        Q[i][j] = T_[0][i]*T_[0][j] + T_[1][i]*T_[1][j] + T_[2][i]*T_[2][j];
    const float mu0 = means3D[e*3+0], mu1 = means3D[e*3+1], mu2 = means3D[e*3+2];
    const float qm0 = Q[0][0]*mu0 + Q[0][1]*mu1 + Q[0][2]*mu2;
    const float qm1 = Q[1][0]*mu0 + Q[1][1]*mu1 + Q[1][2]*mu2;
    const float qm2 = Q[2][0]*mu0 + Q[2][1]*mu1 + Q[2][2]*mu2;
    const float cc  = qm0*mu0 + qm1*mu1 + qm2*mu2 - 1.f;
    // a = d^T Q d
    cA[e*8+0]=Q[0][0]; cA[e*8+1]=Q[1][1]; cA[e*8+2]=Q[2][2];
    cA[e*8+3]=2.f*Q[0][1]; cA[e*8+4]=2.f*Q[0][2]; cA[e*8+5]=2.f*Q[1][2];
    cA[e*8+6]=0.f; cA[e*8+7]=0.f;
    // b = 2 d^T Q o - 2 qm.d
    cB[e*12+0]=2.f*Q[0][0]; cB[e*12+1]=2.f*Q[1][1]; cB[e*12+2]=2.f*Q[2][2];
    cB[e*12+3]=2.f*Q[0][1]; cB[e*12+4]=2.f*Q[0][2]; cB[e*12+5]=2.f*Q[1][2];
    cB[e*12+6]=-2.f*qm0;    cB[e*12+7]=-2.f*qm1;    cB[e*12+8]=-2.f*qm2;
    cB[e*12+9]=0.f; cB[e*12+10]=0.f; cB[e*12+11]=0.f;
    // c = o^T Q o - 2 qm.o + (mu^T Q mu - 1)
    cC[e*12+0]=Q[0][0]; cC[e*12+1]=Q[1][1]; cC[e*12+2]=Q[2][2];
    cC[e*12+3]=2.f*Q[0][1]; cC[e*12+4]=2.f*Q[0][2]; cC[e*12+5]=2.f*Q[1][2];
    cC[e*12+6]=-2.f*qm0;    cC[e*12+7]=-2.f*qm1;    cC[e*12+8]=-2.f*qm2;
    cC[e*12+9]=cc; cC[e*12+10]=0.f; cC[e*12+11]=0.f;
    colOp[e*4+0]=colors[e*3+0]; colOp[e*4+1]=colors[e*3+1];
    colOp[e*4+2]=colors[e*3+2]; colOp[e*4+3]=opacities[e];
  }
  // Per-ray feature vectors (16 rays per block).
  if (tid < RPB) {
    const int m = blockIdx.x * RPB + tid;
    const float ox = rays_o[m*3+0], oy = rays_o[m*3+1], oz = rays_o[m*3+2];
    float dx = rays_d[m*3+0], dy = rays_d[m*3+1], dz = rays_d[m*3+2];
    const float inv = rsqrtf(dx*dx + dy*dy + dz*dz);
    dx *= inv; dy *= inv; dz *= inv;
    fA[0*RPB+tid]=dx*dx; fA[1*RPB+tid]=dy*dy; fA[2*RPB+tid]=dz*dz;
    fA[3*RPB+tid]=dx*dy; fA[4*RPB+tid]=dx*dz; fA[5*RPB+tid]=dy*dz;
    fA[6*RPB+tid]=0.f;   fA[7*RPB+tid]=0.f;
    fB[0*RPB+tid]=dx*ox; fB[1*RPB+tid]=dy*oy; fB[2*RPB+tid]=dz*oz;
    fB[3*RPB+tid]=dx*oy+dy*ox; fB[4*RPB+tid]=dx*oz+dz*ox; fB[5*RPB+tid]=dy*oz+dz*oy;
    fB[6*RPB+tid]=dx; fB[7*RPB+tid]=dy; fB[8*RPB+tid]=dz;
    fB[9*RPB+tid]=0.f; fB[10*RPB+tid]=0.f; fB[11*RPB+tid]=0.f;
    fC[0*RPB+tid]=ox*ox; fC[1*RPB+tid]=oy*oy; fC[2*RPB+tid]=oz*oz;
    fC[3*RPB+tid]=ox*oy; fC[4*RPB+tid]=ox*oz; fC[5*RPB+tid]=oy*oz;
    fC[6*RPB+tid]=ox; fC[7*RPB+tid]=oy; fC[8*RPB+tid]=oz;
    fC[9*RPB+tid]=1.f; fC[10*RPB+tid]=0.f; fC[11*RPB+tid]=0.f;
  }
  __syncthreads();

  // ============ Phase 1: WMMA GEMM -> (a,b,c), then packed events ===========
  {
    const int wave = tid >> 5;          // 8 waves -> 8 tiles of 16 ellipsoids
    const int lane = tid & 31;
    const int row  = lane & 15;
    const int kh   = lane >> 4;
    const int rowBase = wave * 16;

    v8f aT = {0.f,0.f,0.f,0.f,0.f,0.f,0.f,0.f};
    v8f bT = {0.f,0.f,0.f,0.f,0.f,0.f,0.f,0.f};
    v8f cT = {0.f,0.f,0.f,0.f,0.f,0.f,0.f,0.f};
    #pragma unroll
    for (int q = 0; q < 2; ++q) aT = wmma_chunk(cA,  8, fA, rowBase, row, kh, q, aT);
    #pragma unroll
    for (int q = 0; q < 3; ++q) bT = wmma_chunk(cB, 12, fB, rowBase, row, kh, q, bT);
    #pragma unroll
    for (int q = 0; q < 3; ++q) cT = wmma_chunk(cC, 12, fC, rowBase, row, kh, q, cT);

    // Each lane owns 8 (ellipsoid, ray) results of the C tile.
    #pragma unroll
    for (int i = 0; i < 8; ++i) {
      const int ell = rowBase + i + 8 * kh;
      const int ray = row;
      const float a = aT[i], b = bT[i], c = cT[i];
      const float disc = b*b - 4.f*a*c;
      const bool valid = disc >= 0.f;
      const float sq = __builtin_amdgcn_sqrtf(fmaxf(disc, 0.f));
      const float inv2a = 0.5f * __builtin_amdgcn_rcpf(a);   // a = |T d|^2 > 0
      const float t0 = (-b - sq) * inv2a;
      const float t1 = (-b + sq) * inv2a;
      const bool h0 = valid && (t0 > 0.f);
      const bool h1 = valid && (t1 > 0.f);
      // Packed event: ordered-float key (hi 32) | payload (id<<2 | io+1) (lo 32).
      evK[ray*NEV + ell] =
          ((u64)okey(h0 ? t0 : -1.f) << 32) | (unsigned)((ell << 2) | (h0 ? 2u : 1u));
      evK[ray*NEV + N_ELL + ell] =
          ((u64)okey(h1 ? t1 : -1.f) << 32) | (unsigned)((ell << 2) | (h1 ? 0u : 1u));
    }
  }

  // ========= Phase 2: branchless bitonic sort (16 rays in parallel) =========
  for (int size = 2; size <= NEV; size <<= 1) {
    for (int stride = size >> 1; stride >= 1; stride >>= 1) {
      __syncthreads();
      #pragma unroll
      for (int p8 = 0; p8 < 8; ++p8) {
        const int pg = tid + NTHR * p8;        // 0..2047 : 16 rays x 128 pairs
        const int r  = pg >> 7;
        const int p  = pg & 127;
        const int i  = ((p & ~(stride - 1)) << 1) | (p & (stride - 1));
        const int j  = i | stride;
        const bool asc = (i & size) == 0;
        const u64 a = evK[r*NEV + i];
        const u64 b = evK[r*NEV + j];
        const bool sw = asc ? (a > b) : (a < b);
        evK[r*NEV + i] = sw ? b : a;
        evK[r*NEV + j] = sw ? a : b;
      }
    }
  }
  __syncthreads();

  // ---- in-place unpack: evK (32KB) -> evT (low 16KB) + evP, freeing G ------
  {
    u64 v[RPB];
    #pragma unroll
    for (int r = 0; r < RPB; ++r) v[r] = evK[r*NEV + tid];
    __syncthreads();
    #pragma unroll
    for (int r = 0; r < RPB; ++r) {
      evT[r*NEV + tid] = unokey((unsigned)(v[r] >> 32));
      evP[r*NEV + tid] = (unsigned short)(v[r] & 0xFFFFu);
    }
    __syncthreads();
  }

  // ======================= Phase 3: integration =============================
  // k = tid; each thread owns event index k for every ray r.
  // 3a: delta_density -> dens (aliases dead coefficient region), then cumsum.
  #pragma unroll
  for (int r = 0; r < RPB; ++r) {
    float dd = 0.f;
    if (tid < NEV - 1) {
      const unsigned p = evP[r*NEV + tid];
      const int id = (int)(p >> 2);
      const float io = (float)((int)(p & 3) - 1);
      dd = colOp[id*4 + 3] * io;
    }
    dens[r*NEV + tid] = dd;
  }
  __syncthreads();
  for (int off = 1; off < NEV; off <<= 1) {
    float v[RPB];
    #pragma unroll
    for (int r = 0; r < RPB; ++r)
      v[r] = dens[r*NEV + tid] + (tid >= off ? dens[r*NEV + tid - off] : 0.f);
    __syncthreads();
    #pragma unroll
    for (int r = 0; r < RPB; ++r) dens[r*NEV + tid] = v[r];
    __syncthreads();
  }

  // 3b: g = delta_t * density, cumsum -> G (log transmittance).
  #pragma unroll
  for (int r = 0; r < RPB; ++r) {
    float g = 0.f;
    if (tid < NEV - 1)
      g = (evT[r*NEV + tid + 1] - evT[r*NEV + tid]) * dens[r*NEV + tid];
    G[r*NEV + tid] = g;
  }
  __syncthreads();
  for (int off = 1; off < NEV; off <<= 1) {
    float v[RPB];
    #pragma unroll
    for (int r = 0; r < RPB; ++r)
      v[r] = G[r*NEV + tid] + (tid >= off ? G[r*NEV + tid - off] : 0.f);
    __syncthreads();
    #pragma unroll
    for (int r = 0; r < RPB; ++r) G[r*NEV + tid] = v[r];
    __syncthreads();
  }

  // 3c: weight u[k] = prod[k]*tmp[k]/max(density,eps); overwrite G with u.
  //     prod[0]=1 -> E[0]=g[0]=G[0];  k>=1 -> E = G[k]+g[k] = 2G[k]-G[k-1].
  {
    float u[RPB];
    #pragma unroll
    for (int r = 0; r < RPB; ++r) {
      const float Gk = G[r*NEV + tid];
      const float Gm = (tid > 0) ? G[r*NEV + tid - 1] : 0.f;
      const float E  = (tid == 0) ? Gk : (2.f*Gk - Gm);
      u[r] = (tid < NEV - 1)
               ? __expf(-E) * __builtin_amdgcn_rcpf(fmaxf(dens[r*NEV + tid], 1e-6f))
               : 0.f;
    }
    __syncthreads();
    #pragma unroll
    for (int r = 0; r < RPB; ++r) G[r*NEV + tid] = u[r];
    __syncthreads();
  }

  // 3d: suffix-scan u -> U  (folds the color prefix-sum: out = sum_j q_j * U_j).
  for (int off = 1; off < NEV; off <<= 1) {
    float v[RPB];
    #pragma unroll
    for (int r = 0; r < RPB; ++r)
      v[r] = G[r*NEV + tid] + (tid + off < NEV ? G[r*NEV + tid + off] : 0.f);
    __syncthreads();
    #pragma unroll
    for (int r = 0; r < RPB; ++r) G[r*NEV + tid] = v[r];
    __syncthreads();
  }

  // 3e: out[c] = sum_k color[id_k][c]*opac[id_k]*io_k * U[k]  (evT = scratch).
  for (int ch = 0; ch < 3; ++ch) {
    #pragma unroll
    for (int r = 0; r < RPB; ++r) {
      float val = 0.f;
      if (tid < NEV - 1) {
        const unsigned p = evP[r*NEV + tid];
        const int id = (int)(p >> 2);
        const float io = (float)((int)(p & 3) - 1);
        val = colOp[id*4 + ch] * colOp[id*4 + 3] * io * G[r*NEV + tid];
      }
      evT[r*NEV + tid] = val;
    }
    __syncthreads();
    for (int off = 128; off >= 1; off >>= 1) {
      if (tid < off) {
        #pragma unroll
        for (int r = 0; r < RPB; ++r)
          evT[r*NEV + tid] += evT[r*NEV + tid + off];
      }
      __syncthreads();
    }
    if (tid < RPB)
      out[(blockIdx.x * RPB + tid) * 3 + ch] = evT[tid * NEV];
    __syncthreads();
  }
}

extern "C" void kernel_launch(void* const* d_in, const int* in_sizes, int n_in,
                              void* d_out, int out_size, void* d_ws, size_t ws_size,
                              hipStream_t stream) {
  const float* means3D   = (const float*)d_in[0];
  const float* rays_o    = (const float*)d_in[1];
  const float* rays_d    = (const float*)d_in[2];
  const float* rotations = (const float*)d_in[3];
  const float* scales    = (const float*)d_in[4];
  const float* colors    = (const float*)d_in[5];
  const float* opacities = (const float*)d_in[6];
  // d_in[7..9] = K, R, t : unused by the reference math.
  float* out = (float*)d_out;

  const int M = in_sizes[1] / 3;     // number of rays (65536)
  const int grid = M / RPB;          // 4096 blocks, 16 rays each

  ellipsoid_render_kernel<<<grid, NTHR, 0, stream>>>(
      means3D, rays_o, rays_d, rotations, scales, colors, opacities, out);
}